// DualHeadGAT_1915555414047
// MI455X (gfx1250) — compile-verified
//
#include <hip/hip_runtime.h>
#include <hip/hip_bf16.h>

#define N_NODES 50000
#define N_EDGES 800000
#define ETOT    850000   // E + N self-loops
#define DIM     256
#define NEG_SLOPE 0.2f

typedef float v2f __attribute__((ext_vector_type(2)));
typedef float v8f __attribute__((ext_vector_type(8)));

// ---------------------------------------------------------------- utilities

__device__ __forceinline__ float wave_sum32(float v) {
#pragma unroll
    for (int m = 16; m >= 1; m >>= 1) v += __shfl_xor(v, m, 32);
    return v;
}

// float atomic max via int/uint punning (valid with -inf initialization)
__device__ __forceinline__ void atomicMaxF(float* addr, float val) {
    if (val >= 0.0f) atomicMax((int*)addr, __float_as_int(val));
    else             atomicMin((unsigned int*)addr, __float_as_uint(val));
}

// LDS byte offset of a generic pointer to __shared__ (low 32 bits of flat addr:
// shared aperture is selected purely by ADDR[63:32])
__device__ __forceinline__ uint32_t lds_off(const void* p) {
    return (uint32_t)(uintptr_t)p;
}

// async copy 16B global -> LDS (per-lane addresses), tracked by ASYNCcnt
__device__ __forceinline__ void async_b128(uint32_t lds, const void* g) {
    asm volatile("global_load_async_to_lds_b128 %0, %1, off"
                 :: "v"(lds), "v"((uint64_t)(uintptr_t)g) : "memory");
}
__device__ __forceinline__ void async_b128_off16(uint32_t lds, const void* g) {
    asm volatile("global_load_async_to_lds_b128 %0, %1, off offset:16"
                 :: "v"(lds), "v"((uint64_t)(uintptr_t)g) : "memory");
}
__device__ __forceinline__ void wait_async0() {
    asm volatile("s_wait_asynccnt 0" ::: "memory");
}

__global__ void fill_kernel(float* __restrict__ p, float v, size_t n) {
    size_t i = (size_t)blockIdx.x * blockDim.x + threadIdx.x;
    size_t stride = (size_t)gridDim.x * blockDim.x;
    for (; i < n; i += stride) p[i] = v;
}

// ------------------------------------------------- WMMA fp32 GEMM, K=Dout=256
// C[M,256] = A[M,256] @ B[256,256] via V_WMMA_F32_16X16X4_F32.
// Block = 256 thr (8 waves) -> 16 rows x 128 cols; grid = (2, M/16).
// A tile (16x256, 16KB) async-staged to LDS once; B double-buffered in
// 16-krow x 128-col chunks (8KB) via GLOBAL_LOAD_ASYNC_TO_LDS_B128,
// prefetch of chunk kb+1 overlapping the 4 WMMAs on chunk kb.
__global__ __launch_bounds__(256)
void gemm_n256(const float* __restrict__ A, const float* __restrict__ B,
               float* __restrict__ C, int M) {
    __shared__ float lA[16 * 256];       // 16 KB
    __shared__ float lB[2][16 * 128];    // 2 x 8 KB

    const int t    = threadIdx.x;
    const int lane = t & 31;
    const int wave = t >> 5;
    const int row0 = blockIdx.y * 16;
    const int colB = blockIdx.x * 128;   // block column base
    const int wcol = wave * 16;          // wave column offset in block
    if (row0 >= M) return;

    // ---- stage A tile: thread t copies 64B of row (t>>4) ----
    {
        const int r  = t >> 4;            // 0..15
        const int cb = (t & 15) * 64;     // byte offset in 1KB row
        const char* g = (const char*)(A + (size_t)(row0 + r) * DIM) + cb;
        uint32_t l = lds_off((const char*)lA + r * 1024 + cb);
        async_b128(l, g);
        async_b128_off16(l, g);
        async_b128(l + 32, g + 32);
        async_b128_off16(l + 32, g + 32);
    }
    // ---- B chunk loader: chunk kb = B rows [16kb,16kb+16) x cols [colB,colB+128) ----
    auto loadB = [&](int kb, int buf) {
        const int r  = t >> 4;            // chunk row 0..15
        const int cb = (t & 15) * 32;     // byte offset in 512B row
        const char* g = (const char*)(B + (size_t)(kb * 16 + r) * DIM + colB) + cb;
        uint32_t l = lds_off((const char*)&lB[buf][0] + r * 512 + cb);
        async_b128(l, g);
        async_b128_off16(l, g);
    };
    loadB(0, 0);
    wait_async0();
    __syncthreads();

    const int half = lane >> 4;   // 0 -> K+0/1 ; 1 -> K+2/3  (ISA 16x4 A layout)
    const int l16  = lane & 15;

    const float* Af = lA + l16 * 256 + 2 * half;   // + k
    v8f acc = {0.f, 0.f, 0.f, 0.f, 0.f, 0.f, 0.f, 0.f};

    for (int kb = 0; kb < 16; ++kb) {
        const int buf = kb & 1;
        if (kb < 15) loadB(kb + 1, buf ^ 1);       // overlaps compute below
        const float* Bf = &lB[buf][(2 * half) * 128 + wcol + l16];
#pragma unroll
        for (int kk = 0; kk < 16; kk += 4) {
            float2 av = *(const float2*)(Af + kb * 16 + kk);  // K = .. + 2*half
            v2f a; a.x = av.x; a.y = av.y;
            v2f b;
            b.x = Bf[(kk + 0) * 128];              // K row kk + 2*half
            b.y = Bf[(kk + 1) * 128];              // K row kk + 2*half + 1
            acc = __builtin_amdgcn_wmma_f32_16x16x4_f32(
                false, a, false, b, (short)0, acc, false, false);
        }
        wait_async0();
        __syncthreads();
    }

    // D layout: VGPR r -> row = row0 + r + 8*half, col
    float* Crow = C + (size_t)(row0 + half * 8) * DIM + (colB + wcol + l16);
#pragma unroll
    for (int r = 0; r < 8; ++r) Crow[(size_t)r * DIM] = acc[r];
}

// --------------------------------------------- per-node head scores (s_src/s_dst)
// lane l owns channels [8l, 8l+8) -> head h = l>>3; reduce over 8-lane groups.
__global__ __launch_bounds__(256)
void scores_kernel(const float* __restrict__ xw,
                   const float* __restrict__ a_src, const float* __restrict__ a_dst,
                   float* __restrict__ s_src, float* __restrict__ s_dst) {
    const int lane = threadIdx.x & 31;
    const int node = blockIdx.x * 8 + (threadIdx.x >> 5);
    if (node >= N_NODES) return;

    const float4* xr = (const float4*)(xw + (size_t)node * DIM + lane * 8);
    const float4* as = (const float4*)(a_src + lane * 8);
    const float4* ad = (const float4*)(a_dst + lane * 8);
    float4 x0 = xr[0], x1 = xr[1];
    float4 s0 = as[0], s1 = as[1];
    float4 d0 = ad[0], d1 = ad[1];

    float ss = x0.x*s0.x + x0.y*s0.y + x0.z*s0.z + x0.w*s0.w
             + x1.x*s1.x + x1.y*s1.y + x1.z*s1.z + x1.w*s1.w;
    float sd = x0.x*d0.x + x0.y*d0.y + x0.z*d0.z + x0.w*d0.w
             + x1.x*d1.x + x1.y*d1.y + x1.z*d1.z + x1.w*d1.w;

    ss += __shfl_xor(ss, 1, 32); ss += __shfl_xor(ss, 2, 32); ss += __shfl_xor(ss, 4, 32);
    sd += __shfl_xor(sd, 1, 32); sd += __shfl_xor(sd, 2, 32); sd += __shfl_xor(sd, 4, 32);

    if ((lane & 7) == 0) {
        s_src[node * 4 + (lane >> 3)] = ss;
        s_dst[node * 4 + (lane >> 3)] = sd;
    }
}

// ------------------------------------------------- edge softmax stages
__global__ void edge_logits_kernel(const int* __restrict__ ei,
                                   const float* __restrict__ s_src,
                                   const float* __restrict__ s_dst,
                                   float* __restrict__ logits,
                                   float* __restrict__ m) {
    int e = blockIdx.x * blockDim.x + threadIdx.x;
    if (e >= ETOT) return;
    int r = (e < N_EDGES) ? ei[e]            : (e - N_EDGES);
    int c = (e < N_EDGES) ? ei[N_EDGES + e]  : (e - N_EDGES);
#pragma unroll
    for (int h = 0; h < 4; ++h) {
        float v = s_src[r * 4 + h] + s_dst[c * 4 + h];
        v = (v > 0.0f) ? v : NEG_SLOPE * v;           // LeakyReLU
        logits[(size_t)e * 4 + h] = v;
        atomicMaxF(&m[c * 4 + h], v);
    }
}

__global__ void edge_exp_kernel(const int* __restrict__ ei,
                                const float* __restrict__ m,
                                float* __restrict__ exb,      // in: logits, out: exp
                                float* __restrict__ den) {
    int e = blockIdx.x * blockDim.x + threadIdx.x;
    if (e >= ETOT) return;
    int c = (e < N_EDGES) ? ei[N_EDGES + e] : (e - N_EDGES);
#pragma unroll
    for (int h = 0; h < 4; ++h) {
        float x = __expf(exb[(size_t)e * 4 + h] - m[c * 4 + h]);
        exb[(size_t)e * 4 + h] = x;
        atomicAdd(&den[c * 4 + h], x);
    }
}

__global__ void edge_alpha_kernel(const int* __restrict__ ei,
                                  const float* __restrict__ exb,
                                  const float* __restrict__ den,
                                  float* __restrict__ alpha,
                                  float* __restrict__ attn_out) {
    int e = blockIdx.x * blockDim.x + threadIdx.x;
    if (e >= ETOT) return;
    int c = (e < N_EDGES) ? ei[N_EDGES + e] : (e - N_EDGES);
    float s = 0.0f;
#pragma unroll
    for (int h = 0; h < 4; ++h) {
        float a = exb[(size_t)e * 4 + h] / den[c * 4 + h];
        alpha[(size_t)e * 4 + h] = a;
        s += a;
    }
    attn_out[e] += s * 0.125f;   // mean over 4 heads, then /2 across 2 layers
}

// ---------------------------------------------- alpha-weighted scatter aggregate
// one wave per edge; lane l owns channels [8l,8l+8) -> head l>>3
__global__ __launch_bounds__(256)
void aggregate_kernel(const int* __restrict__ ei,
                      const float* __restrict__ alpha,
                      const float* __restrict__ xw,
                      float* __restrict__ agg) {
    const int lane = threadIdx.x & 31;
    const int e = blockIdx.x * 8 + (threadIdx.x >> 5);
    if (e >= ETOT) return;
    int r = (e < N_EDGES) ? ei[e]           : (e - N_EDGES);
    int c = (e < N_EDGES) ? ei[N_EDGES + e] : (e - N_EDGES);
    float a = alpha[(size_t)e * 4 + (lane >> 3)];
    const float4* xr = (const float4*)(xw + (size_t)r * DIM + lane * 8);
    float4 x0 = xr[0], x1 = xr[1];
    float* o = agg + (size_t)c * DIM + lane * 8;
    atomicAdd(o + 0, a * x0.x); atomicAdd(o + 1, a * x0.y);
    atomicAdd(o + 2, a * x0.z); atomicAdd(o + 3, a * x0.w);
    atomicAdd(o + 4, a * x1.x); atomicAdd(o + 5, a * x1.y);
    atomicAdd(o + 6, a * x1.z); atomicAdd(o + 7, a * x1.w);
}

__global__ void finalize_kernel(const float* __restrict__ agg,
                                const float* __restrict__ bias,
                                float* __restrict__ hout) {
    size_t i = (size_t)blockIdx.x * 256 + threadIdx.x;   // N*DIM threads
    float v = agg[i] + bias[i & (DIM - 1)];
    hout[i] = (v > 0.0f) ? v : (__expf(v) - 1.0f);        // ELU
}

// ------------------------------------------------- output heads
__global__ __launch_bounds__(256)
void node_pred_kernel(const float* __restrict__ h, const float* __restrict__ Wn,
                      const float* __restrict__ bn, float* __restrict__ out) {
    const int lane = threadIdx.x & 31;
    const int n = blockIdx.x * 8 + (threadIdx.x >> 5);
    if (n >= N_NODES) return;
    float s = 0.0f;
    for (int k = lane; k < DIM; k += 32) s += h[(size_t)n * DIM + k] * Wn[k];
    s = wave_sum32(s);
    if (lane == 0) out[n] = s + bn[0];
}

__global__ __launch_bounds__(256)
void edge_pred_kernel(const int* __restrict__ ei, const float* __restrict__ Z,
                      const float* __restrict__ h, const float* __restrict__ bb,
                      float* __restrict__ out) {
    const int lane = threadIdx.x & 31;
    const int e = blockIdx.x * 8 + (threadIdx.x >> 5);
    if (e >= N_EDGES) return;
    int r = ei[e], c = ei[N_EDGES + e];
    const float* z0 = Z + (size_t)r * DIM;
    const float* z1 = z0 + (size_t)N_NODES * DIM;
    const float* z2 = z1 + (size_t)N_NODES * DIM;
    const float* hc = h + (size_t)c * DIM;
    float s0 = 0.f, s1 = 0.f, s2 = 0.f;
    for (int k = lane; k < DIM; k += 32) {
        float hv = hc[k];
        s0 += z0[k] * hv; s1 += z1[k] * hv; s2 += z2[k] * hv;
    }
    s0 = wave_sum32(s0); s1 = wave_sum32(s1); s2 = wave_sum32(s2);
    if (lane == 0) {
        out[(size_t)e * 3 + 0] = s0 + bb[0];
        out[(size_t)e * 3 + 1] = s1 + bb[1];
        out[(size_t)e * 3 + 2] = s2 + bb[2];
    }
}

// ---------------------------------------------------------------- launch

extern "C" void kernel_launch(void* const* d_in, const int* in_sizes, int n_in,
                              void* d_out, int out_size, void* d_ws, size_t ws_size,
                              hipStream_t stream) {
    const float* x   = (const float*)d_in[0];
    const int*   ei  = (const int*)  d_in[1];   // [2,E]
    const float* W1  = (const float*)d_in[2];
    const float* a1s = (const float*)d_in[3];
    const float* a1d = (const float*)d_in[4];
    const float* b1  = (const float*)d_in[5];
    const float* W2  = (const float*)d_in[6];
    const float* a2s = (const float*)d_in[7];
    const float* a2d = (const float*)d_in[8];
    const float* b2  = (const float*)d_in[9];
    const float* Wn  = (const float*)d_in[10];
    const float* bn  = (const float*)d_in[11];
    const float* Wb  = (const float*)d_in[12];  // [3,256,256]
    const float* bb  = (const float*)d_in[13];

    float* out      = (float*)d_out;
    float* node_out = out;                                    // [N]
    float* edge_out = out + N_NODES;                          // [E,3]
    float* h2       = edge_out + (size_t)3 * N_EDGES;         // [N,256]
    float* attn_out = h2 + (size_t)N_NODES * DIM;             // [ETOT]

    float* ws = (float*)d_ws;
    const size_t NF = (size_t)N_NODES * DIM;
    float* xw   = ws;            // [N,256]
    float* agg  = ws + NF;       // [N,256]
    float* h1   = ws + 2 * NF;   // [N,256]
    float* ssrc = ws + 3 * NF;                         // [N,4]
    float* sdst = ssrc + (size_t)N_NODES * 4;
    float* mbuf = sdst + (size_t)N_NODES * 4;
    float* den  = mbuf + (size_t)N_NODES * 4;
    float* exb  = den  + (size_t)N_NODES * 4;          // [ETOT,4]
    float* alph = exb  + (size_t)ETOT * 4;             // [ETOT,4]
    float* Z    = ws;  // overlaps xw/agg/h1 — all dead before Z GEMMs run

    const dim3 gGemm(2, N_NODES / 16), bGemm(256);
    const int gEdgeT = (ETOT + 255) / 256;             // thread-per-edge kernels
    const int gEdgeW = (ETOT + 7) / 8;                 // wave-per-edge kernels
    const int gNodeW = (N_NODES + 7) / 8;
    const float NEG_INF = -__builtin_inff();

    fill_kernel<<<1024, 256, 0, stream>>>(attn_out, 0.0f, (size_t)ETOT);

    for (int layer = 0; layer < 2; ++layer) {
        const float* in  = (layer == 0) ? x   : h1;
        const float* W   = (layer == 0) ? W1  : W2;
        const float* as  = (layer == 0) ? a1s : a2s;
        const float* ad  = (layer == 0) ? a1d : a2d;
        const float* bia = (layer == 0) ? b1  : b2;
        float* hout      = (layer == 0) ? h1  : h2;

        fill_kernel<<<512, 256, 0, stream>>>(mbuf, NEG_INF, (size_t)N_NODES * 4);
        fill_kernel<<<512, 256, 0, stream>>>(den, 0.0f, (size_t)N_NODES * 4);
        fill_kernel<<<2048, 256, 0, stream>>>(agg, 0.0f, NF);

        gemm_n256<<<gGemm, bGemm, 0, stream>>>(in, W, xw, N_NODES);
        scores_kernel<<<gNodeW, 256, 0, stream>>>(xw, as, ad, ssrc, sdst);
        edge_logits_kernel<<<gEdgeT, 256, 0, stream>>>(ei, ssrc, sdst, exb, mbuf);
        edge_exp_kernel<<<gEdgeT, 256, 0, stream>>>(ei, mbuf, exb, den);
        edge_alpha_kernel<<<gEdgeT, 256, 0, stream>>>(ei, exb, den, alph, attn_out);
        aggregate_kernel<<<gEdgeW, 256, 0, stream>>>(ei, alph, xw, agg);
        finalize_kernel<<<(int)(NF / 256), 256, 0, stream>>>(agg, bia, hout);
    }

    // Bilinear edge head: Z_o = h2 @ Wb[o]  (three more WMMA GEMMs)
    for (int o = 0; o < 3; ++o)
        gemm_n256<<<gGemm, bGemm, 0, stream>>>(h2, Wb + (size_t)o * DIM * DIM,
                                               Z + (size_t)o * NF, N_NODES);

    node_pred_kernel<<<gNodeW, 256, 0, stream>>>(h2, Wn, bn, node_out);
    edge_pred_kernel<<<(N_EDGES + 7) / 8, 256, 0, stream>>>(ei, Z, h2, bb, edge_out);
}